// PMSN_kernel_63642825392757
// MI455X (gfx1250) — compile-verified
//
#include <hip/hip_runtime.h>
#include <math.h>

// S4D kernel generation on gfx1250 via V_WMMA_F32_16X16X4_F32.
//
//   out[h, 16*lq + lr] = sum_m U[lq, m] * V[m, lr]   (m = 0..7)
//   U[lq, 2n]   =  Re(w_n * exp(Adt_n * 16*lq))      (w = C * B_bar)
//   U[lq, 2n+1] = -Im(w_n * exp(Adt_n * 16*lq))
//   V[2n,  lr]  =  Re(exp(Adt_n * lr))
//   V[2n+1,lr]  =  Im(exp(Adt_n * lr))
//
// One wave handles one (h, block-of-256-l): two chained 16x16x4 f32 WMMAs.

typedef __attribute__((ext_vector_type(2))) float v2f;
typedef __attribute__((ext_vector_type(8))) float v8f;

#define S4D_H 2048
#define S4D_N 4
#define S4D_L 4096

__global__ __launch_bounds__(256) void s4d_wmma_kernel(
    const float* __restrict__ log_dt,
    const float* __restrict__ log_A_real,
    const float* __restrict__ A_imag,
    const float* __restrict__ VinvB_real,
    const float* __restrict__ VinvB_imag,
    const float* __restrict__ CV_real,
    const float* __restrict__ CV_imag,
    float* __restrict__ out)
{
    const int lane = threadIdx.x & 31;
    const int wave = blockIdx.x * 8 + (threadIdx.x >> 5);
    const int h = wave >> 4;        // channel 0..2047 (wave-uniform)
    const int b = wave & 15;        // 256-wide l-block, 0..15

    const int hf  = lane >> 4;      // lane half: selects K-pair / D row-half
    const int row = lane & 15;      // A: M (=lq_local); B: N (=lr); D: N

    // ---- per-h mode parameters (wave-uniform scalar math) ----
    const float dt = expf(log_dt[h]);

    float Adt_r[S4D_N], Adt_i[S4D_N], w_r[S4D_N], w_i[S4D_N];
#pragma unroll
    for (int n = 0; n < S4D_N; ++n) {
        const int idx = h * S4D_N + n;
        const float Ar = -expf(log_A_real[idx]);
        const float Ai = A_imag[idx];
        const float ar = Ar * dt, ai = Ai * dt;     // Adt
        Adt_r[n] = ar; Adt_i[n] = ai;

        // A_bar = exp(Adt)
        float si, co;
        sincosf(ai, &si, &co);
        const float e = expf(ar);
        const float abr = e * co, abi = e * si;

        // B_bar = (A_bar - 1) * B / A
        const float Br = VinvB_real[idx], Bi = VinvB_imag[idx];
        const float tr = abr - 1.0f, ti = abi;
        const float tbr = tr * Br - ti * Bi;
        const float tbi = tr * Bi + ti * Br;
        const float den = Ar * Ar + Ai * Ai;
        const float bbr = (tbr * Ar + tbi * Ai) / den;
        const float bbi = (tbi * Ar - tbr * Ai) / den;

        // w = C * B_bar
        const float Cr = CV_real[idx], Ci = CV_imag[idx];
        w_r[n] = Cr * bbr - Ci * bbi;
        w_i[n] = Cr * bbi + Ci * bbr;
    }

    // ---- A-matrix fragments: (Re, -Im) of w_n * exp(Adt_n * 16*lq) ----
    // global lq = 16*b + row  ->  phase arg s = 256*b + 16*row
    const float sA = (float)(256 * b + 16 * row);
    auto makeA = [&](int n) -> v2f {
        float si, co;
        sincosf(Adt_i[n] * sA, &si, &co);
        const float e  = expf(Adt_r[n] * sA);
        const float pr = e * co, pi = e * si;
        v2f r;
        r.x = w_r[n] * pr - w_i[n] * pi;        //  Re(w * P^lq)
        r.y = -(w_r[n] * pi + w_i[n] * pr);     // -Im(w * P^lq)
        return r;
    };

    // ---- B-matrix fragments: (Re, Im) of exp(Adt_n * lr) ----
    const float sB = (float)row;                // lr
    auto makeB = [&](int n) -> v2f {
        float si, co;
        sincosf(Adt_i[n] * sB, &si, &co);
        const float e = expf(Adt_r[n] * sB);
        v2f r;
        r.x = e * co;                           // Re(Q^lr)
        r.y = e * si;                           // Im(Q^lr)
        return r;
    };

    // Lane-half hf covers mode n = hf for K-chunk 0 and n = 2+hf for chunk 1
    const v2f A0 = makeA(hf),     B0 = makeB(hf);
    const v2f A1 = makeA(2 + hf), B1 = makeB(2 + hf);

    v8f acc = {};
    // D = A0*B0 + A1*B1  (rank-8 contraction as two chained 16x16x4 WMMAs)
    acc = __builtin_amdgcn_wmma_f32_16x16x4_f32(
        false, A0, false, B0, (short)0, acc, false, false);
    acc = __builtin_amdgcn_wmma_f32_16x16x4_f32(
        false, A1, false, B1, (short)0, acc, false, false);

    // ---- D layout store: VGPR r, lane -> M = r + 8*hf, N = row ----
    // l = 256*b + 16*M + N
    float* o = out + (size_t)h * S4D_L + 256 * b + 128 * hf + row;
#pragma unroll
    for (int r = 0; r < 8; ++r) {
        o[16 * r] = acc[r];
    }
}

extern "C" void kernel_launch(void* const* d_in, const int* in_sizes, int n_in,
                              void* d_out, int out_size, void* d_ws, size_t ws_size,
                              hipStream_t stream) {
    (void)in_sizes; (void)n_in; (void)out_size; (void)d_ws; (void)ws_size;
    const float* log_dt     = (const float*)d_in[0];
    const float* log_A_real = (const float*)d_in[1];
    const float* A_imag     = (const float*)d_in[2];
    const float* VinvB_real = (const float*)d_in[3];
    const float* VinvB_imag = (const float*)d_in[4];
    const float* CV_real    = (const float*)d_in[5];
    const float* CV_imag    = (const float*)d_in[6];
    // d_in[7] is L (scalar); fixed at 4096 by setup_inputs.
    float* out = (float*)d_out;

    // 2048 h * 16 l-blocks = 32768 waves; 8 waves per 256-thread block.
    dim3 grid(S4D_H * 16 / 8);
    dim3 block(256);
    s4d_wmma_kernel<<<grid, block, 0, stream>>>(
        log_dt, log_A_real, A_imag, VinvB_real, VinvB_imag, CV_real, CV_imag, out);
}